// CrossEntropyLabelSmooth_81320910782918
// MI455X (gfx1250) — compile-verified
//
#include <hip/hip_runtime.h>
#include <hip/hip_bf16.h>
#include <stdint.h>

// CrossEntropyLabelSmooth on MI455X (gfx1250).
// Memory-bound streaming reduction: 524 MB read once -> ~22.5 us floor @ 23.3 TB/s.
// One workgroup per row; whole 128,000-byte row streamed into LDS via
// global_load_async_to_lds_b128 (ASYNCcnt, non-temporal), then two LDS passes
// (max + sum, then sum of exp) and LDS-resident gathers. 2 blocks/WGP (320 KB LDS).

#define B_ROWS   4096
#define C_CLS    32000
#define P_POS    50
#define VEC4     (C_CLS / 4)        // 8000 float4 per row
#define TPB      512                // 16 wave32 waves
#define NWAVES   (TPB / 32)
#define FULL_IT  (VEC4 / TPB)       // 15 uniform iterations
#define TAIL_N   (VEC4 - FULL_IT * TPB)   // 320 remaining float4s

#define EPS_F    0.1f
#define SW_F     0.1f
#define LAM_F    0.2f

// gfx1250 async global->LDS load, GVS mode: SGPR64 base + VGPR32 byte offset.
// Non-temporal: row data is consumed once from LDS, keep it out of L2's way.
__device__ __forceinline__ void async_load_b128_to_lds(unsigned lds_byte_addr,
                                                       unsigned gl_byte_off,
                                                       const void* gl_base) {
    asm volatile("global_load_async_to_lds_b128 %0, %1, %2 th:TH_LOAD_NT"
                 :: "v"(lds_byte_addr), "v"(gl_byte_off), "s"(gl_base)
                 : "memory");
}

__device__ __forceinline__ void wait_async_zero() {
    asm volatile("s_wait_asynccnt 0" ::: "memory");
}

__global__ __launch_bounds__(TPB) void row_loss_kernel(
    const float* __restrict__ x,        // [B_ROWS, C_CLS]
    const int*   __restrict__ tgt,      // [B_ROWS]
    const int*   __restrict__ posvid,   // [B_ROWS, P_POS]
    float*       __restrict__ row_out)  // [B_ROWS]
{
    __shared__ __align__(16) float srow[C_CLS];   // 128,000 B: whole row
    __shared__ float red_a[NWAVES];
    __shared__ float red_b[NWAVES];
    __shared__ float s_rowmax;
    __shared__ int   s_idx[P_POS + 1];            // [0]=target, [1..50]=posvid

    const int row  = blockIdx.x;
    const int tid  = threadIdx.x;
    const int lane = tid & 31;
    const int wid  = tid >> 5;

    const float* rx = x + (size_t)row * (size_t)C_CLS;

    // ---- Stage 0: kick off index loads; latency overlaps the row stream ----
    if (tid == 0)    s_idx[0] = tgt[row];
    if (tid < P_POS) s_idx[tid + 1] = posvid[row * P_POS + tid];

    // ---- Stage 1: async-stream the whole row into LDS (single HBM read) ----
    // Uniform fully-unrolled issue loop: no EXEC-mask churn between issues.
    const unsigned lds_base = (unsigned)(uintptr_t)(&srow[0]);
    #pragma unroll
    for (int k = 0; k < FULL_IT; ++k) {
        const unsigned byte_off = (unsigned)((tid + k * TPB) * 16);
        async_load_b128_to_lds(lds_base + byte_off, byte_off, (const void*)rx);
    }
    if (tid < TAIL_N) {
        const unsigned byte_off = (unsigned)((tid + FULL_IT * TPB) * 16);
        async_load_b128_to_lds(lds_base + byte_off, byte_off, (const void*)rx);
    }
    wait_async_zero();      // each wave drains its own ASYNCcnt
    __syncthreads();        // all waves' LDS data (and s_idx) visible to everyone

    // ---- Stage 2, pass A: row max and row sum (for sum of log-probs) ----
    const float4* s4 = (const float4*)srow;
    float lmax = -INFINITY;
    float lsum = 0.0f;
    #pragma unroll
    for (int k = 0; k < FULL_IT; ++k) {
        float4 d = s4[tid + k * TPB];
        lmax = fmaxf(fmaxf(fmaxf(lmax, d.x), fmaxf(d.y, d.z)), d.w);
        lsum += (d.x + d.y) + (d.z + d.w);
    }
    if (tid < TAIL_N) {
        float4 d = s4[tid + FULL_IT * TPB];
        lmax = fmaxf(fmaxf(fmaxf(lmax, d.x), fmaxf(d.y, d.z)), d.w);
        lsum += (d.x + d.y) + (d.z + d.w);
    }
    #pragma unroll
    for (int o = 16; o > 0; o >>= 1) {            // wave32 butterfly
        lmax = fmaxf(lmax, __shfl_xor(lmax, o, 32));
        lsum += __shfl_xor(lsum, o, 32);
    }
    if (lane == 0) { red_a[wid] = lmax; red_b[wid] = lsum; }
    __syncthreads();

    float Tsum = 0.0f;                            // live only in thread 0
    if (tid == 0) {
        float rm = red_a[0]; Tsum = red_b[0];
        #pragma unroll
        for (int w = 1; w < NWAVES; ++w) { rm = fmaxf(rm, red_a[w]); Tsum += red_b[w]; }
        s_rowmax = rm;
    }
    __syncthreads();
    const float rowmax = s_rowmax;

    // ---- Stage 2, pass B: sum of exp(x - rowmax), from LDS ----
    float lse = 0.0f;
    #pragma unroll
    for (int k = 0; k < FULL_IT; ++k) {
        float4 d = s4[tid + k * TPB];
        lse += (__expf(d.x - rowmax) + __expf(d.y - rowmax)) +
               (__expf(d.z - rowmax) + __expf(d.w - rowmax));
    }
    if (tid < TAIL_N) {
        float4 d = s4[tid + FULL_IT * TPB];
        lse += (__expf(d.x - rowmax) + __expf(d.y - rowmax)) +
               (__expf(d.z - rowmax) + __expf(d.w - rowmax));
    }
    #pragma unroll
    for (int o = 16; o > 0; o >>= 1) lse += __shfl_xor(lse, o, 32);
    if (lane == 0) red_a[wid] = lse;              // prior reads fenced by barrier above
    __syncthreads();

    // ---- Stage 3: LDS gathers + closed-form row loss (thread 0) ----
    if (tid == 0) {
        float ssum = 0.0f;
        #pragma unroll
        for (int w = 0; w < NWAVES; ++w) ssum += red_a[w];
        const float logZ = rowmax + __logf(ssum);

        const float xt = srow[s_idx[0]];
        float psum = 0.0f;
        #pragma unroll 10
        for (int p = 0; p < P_POS; ++p) psum += srow[s_idx[p + 1]];

        const float sum_lp = Tsum - (float)C_CLS * logZ;    // sum_c log_probs
        const float lp_t   = xt - logZ;                     // log_prob[target]
        const float lp_pos = psum - (float)P_POS * logZ;    // sum_p log_prob[pos]

        const float hard = -(1.0f - EPS_F) * lp_t - (EPS_F / (float)C_CLS) * sum_lp;
        const float soft = -(1.0f - LAM_F) * lp_t - (LAM_F / (float)P_POS) * lp_pos;
        row_out[row] = (1.0f - SW_F) * hard + SW_F * soft;
    }
}

__global__ __launch_bounds__(256) void final_reduce_kernel(
    const float* __restrict__ rows, float* __restrict__ out)
{
    __shared__ float red[8];
    float s = 0.0f;
    for (int i = threadIdx.x; i < B_ROWS; i += 256) s += rows[i];
    #pragma unroll
    for (int o = 16; o > 0; o >>= 1) s += __shfl_xor(s, o, 32);
    if ((threadIdx.x & 31) == 0) red[threadIdx.x >> 5] = s;
    __syncthreads();
    if (threadIdx.x == 0) {
        float t = 0.0f;
        #pragma unroll
        for (int w = 0; w < 8; ++w) t += red[w];
        out[0] = t / (float)B_ROWS;   // mean over rows
    }
}

extern "C" void kernel_launch(void* const* d_in, const int* in_sizes, int n_in,
                              void* d_out, int out_size, void* d_ws, size_t ws_size,
                              hipStream_t stream) {
    const float* x   = (const float*)d_in[0];   // [4096, 32000] f32
    const int*   tgt = (const int*)d_in[1];     // [4096] int
    const int*   pv  = (const int*)d_in[2];     // [4096, 50] int
    float* rowbuf = (float*)d_ws;               // 4096 floats of scratch

    row_loss_kernel<<<B_ROWS, TPB, 0, stream>>>(x, tgt, pv, rowbuf);
    final_reduce_kernel<<<1, 256, 0, stream>>>(rowbuf, (float*)d_out);
}